// WaveletTransform_32710470926738
// MI455X (gfx1250) — compile-verified
//
#include <hip/hip_runtime.h>

typedef float v2f __attribute__((ext_vector_type(2)));
typedef float v8f __attribute__((ext_vector_type(8)));

#define N_IMG   1024
#define OUT_N   513          // (1024 + 3) / 2
#define TILE_R  16           // output rows per WG
#define TILE_C  64           // output cols per WG
#define IN_ROWS 34           // 2*16 + 2 halo rows
#define IN_COLS 132          // 2*64 + 4 halo cols
#define PITCH   132

// symmetric reflect (pywt 'symmetric'): -1 -> 0, -2 -> 1, N -> N-1, N+1 -> N-2
__device__ __forceinline__ int mir(int i, int n) {
    i = (i < 0) ? (-i - 1) : i;
    i = (i >= n) ? (2 * n - 1 - i) : i;
    i = (i < 0) ? 0 : i;             // clamp far OOB (don't-care lanes of edge tiles)
    i = (i >= n) ? (n - 1) : i;
    return i;
}

// reversed DEC_LO (true convolution => filter flipped) for the horizontal pass
__device__ __forceinline__ float coefB(int d) {
    float r = 0.0f;
    r = (d == 0) ?  0.48296291314469025f : r;
    r = (d == 1) ?  0.836516303737469f   : r;
    r = (d == 2) ?  0.22414386804185735f : r;
    r = (d == 3) ? -0.12940952255092145f : r;
    return r;
}

__launch_bounds__(256)
__global__ void dwt2_fused_wmma(const float* __restrict__ x,
                                float* __restrict__ out,
                                int numImg) {
    __shared__ float lds[IN_ROWS * PITCH + 2 * TILE_R * PITCH];
    float* sIn  = lds;
    float* sVlo = lds + IN_ROWS * PITCH;
    float* sVhi = sVlo + TILE_R * PITCH;

    const int tid = threadIdx.x;
    const int c0  = blockIdx.x * TILE_C;
    const int r0  = blockIdx.y * TILE_R;
    const int b   = blockIdx.z;
    const float* img = x + (size_t)b * N_IMG * N_IMG;

    // ---------- load 34 x 132 mirrored input tile (coalesced along columns) ----------
    const int gr0 = 2 * r0 - 2;
    const int gc0 = 2 * c0 - 2;
    for (int p = tid; p < IN_ROWS * IN_COLS; p += 256) {
        int row = p / IN_COLS;
        int col = p - row * IN_COLS;
        int gr = mir(gr0 + row, N_IMG);
        int gc = mir(gc0 + col, N_IMG);
        sIn[row * PITCH + col] = img[(size_t)gr * N_IMG + gc];
    }
    __syncthreads();

    // ---------- stage 1: vertical 4-tap, stride-2 (lo and hi) ----------
    const float fl0 =  0.48296291314469025f, fl1 =  0.836516303737469f,
                fl2 =  0.22414386804185735f, fl3 = -0.12940952255092145f;   // rev(DEC_LO)
    const float fh0 = -0.12940952255092145f, fh1 = -0.22414386804185735f,
                fh2 =  0.836516303737469f,   fh3 = -0.48296291314469025f;   // rev(DEC_HI)
    for (int p = tid; p < TILE_R * IN_COLS; p += 256) {
        int r = p / IN_COLS;
        int j = p - r * IN_COLS;
        float a0 = sIn[(2 * r + 0) * PITCH + j];
        float a1 = sIn[(2 * r + 1) * PITCH + j];
        float a2 = sIn[(2 * r + 2) * PITCH + j];
        float a3 = sIn[(2 * r + 3) * PITCH + j];
        sVlo[r * PITCH + j] = fl0 * a0 + fl1 * a1 + fl2 * a2 + fl3 * a3;
        sVhi[r * PITCH + j] = fh0 * a0 + fh1 * a1 + fh2 * a2 + fh3 * a3;
    }
    __syncthreads();

    // ---------- stage 2: horizontal lo as banded matmul via V_WMMA_F32_16X16X4_F32 ----
    // Wave w: source = (w<4 ? vlo : vhi), 16x16 output block q = w&3.
    // out[M][n] = sum_s src[M][32q + s] * B[s][n],  B[s][n] = flo_rev[s-2n] (0<=s-2n<=3)
    const int wave   = tid >> 5;
    const int lane   = tid & 31;
    const float* src = (wave < 4) ? sVlo : sVhi;
    const int q      = wave & 3;
    const int laneN  = lane & 15;
    const int hiHalf = lane >> 4;        // A/B VGPR layout: lanes 0-15 K={0,1}, 16-31 K={2,3}
    const int kBase  = hiHalf * 2;

    v8f acc = {};
    #pragma unroll
    for (int t = 0; t < 9; ++t) {
        const float* arow = src + laneN * PITCH + 32 * q + 4 * t + kBase;
        v2f A, B;
        A.x = arow[0];                               // A[M=laneN][K=kBase]
        A.y = arow[1];                               // A[M=laneN][K=kBase+1]
        B.x = coefB(4 * t + kBase + 0 - 2 * laneN);  // B[K=kBase  ][N=laneN]
        B.y = coefB(4 * t + kBase + 1 - 2 * laneN);  // B[K=kBase+1][N=laneN]
        acc = __builtin_amdgcn_wmma_f32_16x16x4_f32(
            false, A, false, B, (short)0, acc, false, false);
    }

    // ---------- store 16x16 f32 D block (VGPR v: rows {v, v+8}), guarded at 513 edges --
    const size_t plane   = (size_t)OUT_N * OUT_N;
    float* outp = out + ((wave < 4) ? (size_t)0 : (size_t)numImg * plane) + (size_t)b * plane;
    const int colO  = c0 + 16 * q + laneN;
    const int mBase = hiHalf * 8;
    #pragma unroll
    for (int v = 0; v < 8; ++v) {
        int rowO = r0 + mBase + v;
        if (rowO < OUT_N && colO < OUT_N)
            outp[(size_t)rowO * OUT_N + colO] = acc[v];
    }
}

extern "C" void kernel_launch(void* const* d_in, const int* in_sizes, int n_in,
                              void* d_out, int out_size, void* d_ws, size_t ws_size,
                              hipStream_t stream) {
    (void)n_in; (void)out_size; (void)d_ws; (void)ws_size;
    const float* x = (const float*)d_in[0];
    float* out = (float*)d_out;
    const int numImg = in_sizes[0] / (N_IMG * N_IMG);   // 16*3 = 48
    dim3 grid((OUT_N + TILE_C - 1) / TILE_C,            // 9
              (OUT_N + TILE_R - 1) / TILE_R,            // 33
              numImg);                                  // 48
    dwt2_fused_wmma<<<grid, 256, 0, stream>>>(x, out, numImg);
}